// Conv2d_35742717837647
// MI455X (gfx1250) — compile-verified
//
#include <hip/hip_runtime.h>
#include <hip/hip_bf16.h>

typedef float v2f __attribute__((ext_vector_type(2)));
typedef float v4f __attribute__((ext_vector_type(4)));
typedef float v8f __attribute__((ext_vector_type(8)));

#define IN_H 512
#define IN_W 512
#define OUT_H 506
#define OUT_W 506
#define KH 7
#define KW 7

// Each block: 8 waves, each wave one 16x16 output tile -> 16(oy) x 128(ox) strip.
#define XROWS 22          // 16 + KH - 1 input rows needed
#define XCOLS 136         // 128 + halo(6), padded to multiple of 8 floats
#define KCHUNKS 6         // ceil((16+KW-1)/4) = 6 chunks of K=4 (24 banded cols)
#define WB_ELEMS (KH * KCHUNKS * 16 * 4)   // [ky][chunk][n][kk] fragment-ordered

__global__ __launch_bounds__(256)
void conv7x7_wmma_f32(const float* __restrict__ in,
                      const float* __restrict__ wgt,
                      const float* __restrict__ bias,
                      float* __restrict__ out)
{
    __shared__ float sX[XROWS * XCOLS];   // input strip (row-major)
    __shared__ float sW[WB_ELEMS];        // banded weights, B-fragment order

    const int tid = threadIdx.x;
    const int b   = blockIdx.z;
    const int oy0 = blockIdx.y * 16;      // output row base of this block
    const int oxs = blockIdx.x * 128;     // output col base of this strip

    // ---- Build banded weight matrices in WMMA-B fragment order ----
    // sW index = ((ky*KCHUNKS + c)*16 + n)*4 + kk ; value = w[ky, (4c+kk) - n] or 0
    for (int i = tid; i < WB_ELEMS; i += 256) {
        const int kk = i & 3;
        const int t1 = i >> 2;
        const int n  = t1 & 15;
        const int t2 = t1 >> 4;
        const int c  = t2 % KCHUNKS;
        const int ky = t2 / KCHUNKS;
        const int kx = (4 * c + kk) - n;
        sW[i] = (kx >= 0 && kx < KW) ? wgt[ky * KW + kx] : 0.0f;
    }

    // ---- Stage input strip into LDS with 128-bit global loads ----
    // Interior float4s are 16B-aligned (oxs % 128 == 0, XCOLS % 4 == 0); only the
    // right-edge strip needs the clamped scalar path. Clamped rows/cols only feed
    // outputs that are masked at the store.
    const float* inb = in + (size_t)b * IN_H * IN_W;
    {
        const int NV = (XROWS * XCOLS) / 4;          // 748 float4 groups
        for (int j = tid; j < NV; j += 256) {
            const int row = j / (XCOLS / 4);
            const int c4  = (j - row * (XCOLS / 4)) * 4;
            int gr = oy0 + row; if (gr > IN_H - 1) gr = IN_H - 1;
            const int gc = oxs + c4;
            v4f v;
            if (gc + 3 <= IN_W - 1) {
                v = *(const v4f*)&inb[gr * IN_W + gc];     // global_load_b128
            } else {
                const float* rowp = &inb[gr * IN_W];
                int g0 = gc;     if (g0 > IN_W - 1) g0 = IN_W - 1;
                int g1 = gc + 1; if (g1 > IN_W - 1) g1 = IN_W - 1;
                int g2 = gc + 2; if (g2 > IN_W - 1) g2 = IN_W - 1;
                int g3 = gc + 3; if (g3 > IN_W - 1) g3 = IN_W - 1;
                v = (v4f){rowp[g0], rowp[g1], rowp[g2], rowp[g3]};
            }
            *(v4f*)&sX[row * XCOLS + c4] = v;              // ds_store_b128
        }
    }
    __syncthreads();

    const int lane = tid & 31;
    const int wv   = tid >> 5;        // wave id 0..7 -> tile within strip
    const int oxt  = wv * 16;         // tile's local column base

    // A-fragment addressing (16x4 f32): lanes 0-15 hold M=lane with K=0,1;
    // lanes 16-31 hold M=lane-16 with K=2,3. One float2 LDS load per fragment.
    const int m  = lane & 15;
    const int hk = (lane >> 4) * 2;   // 0 or 2

    const float bv = bias[0];
    v8f acc;
#pragma unroll
    for (int i = 0; i < 8; ++i) acc[i] = bv;

#pragma unroll
    for (int ky = 0; ky < KH; ++ky) {
#pragma unroll
        for (int c = 0; c < KCHUNKS; ++c) {
            const v2f a = *(const v2f*)&sX[(m + ky) * XCOLS + (oxt + 4 * c + hk)];
            const v2f bm = *(const v2f*)&sW[(((ky * KCHUNKS + c) * 16) + m) * 4 + hk];
            // D = A(16x4) * B(4x16) + C ; full-rate fp32 matrix path on CDNA5
            acc = __builtin_amdgcn_wmma_f32_16x16x4_f32(
                false, a, false, bm, (short)0, acc, false, false);
        }
    }

    // ---- Store: C/D layout = 8 VGPRs; lane<16 -> M=v, lane>=16 -> M=v+8; N=lane&15
    const int n  = lane & 15;
    const int mo = (lane >> 4) * 8;
    const int ox = oxs + oxt + n;
    if (ox < OUT_W) {
#pragma unroll
        for (int v = 0; v < 8; ++v) {
            const int oy = oy0 + mo + v;
            if (oy < OUT_H)
                out[((size_t)b * OUT_H + oy) * OUT_W + ox] = acc[v];
        }
    }
}

extern "C" void kernel_launch(void* const* d_in, const int* in_sizes, int n_in,
                              void* d_out, int out_size, void* d_ws, size_t ws_size,
                              hipStream_t stream) {
    const float* enc_x  = (const float*)d_in[0];   // (64,1,512,512) fp32
    const float* weight = (const float*)d_in[1];   // (1,1,7,7) fp32
    const float* bias   = (const float*)d_in[2];   // (1,) fp32
    float* out = (float*)d_out;                    // (64,1,506,506) fp32

    dim3 block(256, 1, 1);
    dim3 grid((OUT_W + 127) / 128,   // 4 strips of 128 output cols
              (OUT_H + 15) / 16,     // 32 row-tiles
              64);                   // batch
    conv7x7_wmma_f32<<<grid, block, 0, stream>>>(enc_x, weight, bias, out);
}